// DirSageConv_27152783245350
// MI455X (gfx1250) — compile-verified
//
#include <hip/hip_runtime.h>

#define D 64
#define ALPHA 0.5f

typedef __attribute__((ext_vector_type(2))) float v2f;
typedef __attribute__((ext_vector_type(4))) float v4f;
typedef __attribute__((ext_vector_type(8))) float v8f;

// ---------------------------------------------------------------------------
// Kernel 1: zero the workspace accumulators (sum_in, sum_out, cnt_in, cnt_out)
// ---------------------------------------------------------------------------
__global__ void zero_ws_kernel(float* __restrict__ ws, int n) {
    int i = blockIdx.x * blockDim.x + threadIdx.x;
    if (i < n) ws[i] = 0.0f;
}

// ---------------------------------------------------------------------------
// Kernel 2: edge scatter. One thread per (edge, 4-feature chunk):
// b128 gathers (16 threads span a full 256B row) + f32 atomics into the
// L2-resident accumulators (x=25.6MB, sums=51MB, all < 192MB L2).
// ---------------------------------------------------------------------------
__global__ void scatter_edges_kernel(const float* __restrict__ x,
                                     const int*   __restrict__ ei,
                                     float* __restrict__ sum_in,
                                     float* __restrict__ sum_out,
                                     float* __restrict__ cnt_in,
                                     float* __restrict__ cnt_out,
                                     int E) {
    long long tid = (long long)blockIdx.x * blockDim.x + threadIdx.x;
    long long tot = (long long)E * (D / 4);
    if (tid >= tot) return;
    int e = (int)(tid >> 4);         // / 16 chunks per edge
    int f = (int)(tid & 15) * 4;     // starting feature of this chunk
    int s = ei[e];                   // src
    int d = ei[E + e];               // dst

    v4f xs = *(const v4f*)(x + (long long)s * D + f);
    v4f xd = *(const v4f*)(x + (long long)d * D + f);

    float* pin  = sum_in  + (long long)d * D + f;
    float* pout = sum_out + (long long)s * D + f;
    // flow source->target: x[src] aggregated at dst
    atomicAdd(pin + 0, xs.x); atomicAdd(pin + 1, xs.y);
    atomicAdd(pin + 2, xs.z); atomicAdd(pin + 3, xs.w);
    // flow target->source: x[dst] aggregated at src
    atomicAdd(pout + 0, xd.x); atomicAdd(pout + 1, xd.y);
    atomicAdd(pout + 2, xd.z); atomicAdd(pout + 3, xd.w);

    if (f == 0) {
        atomicAdd(&cnt_in[d], 1.0f);
        atomicAdd(&cnt_out[s], 1.0f);
    }
}

// ---------------------------------------------------------------------------
// Kernel 3: fused normalize + 3-way GEMM + bias using V_WMMA_F32_16X16X4_F32.
// One wave computes a 16x64 output tile. Weights staged in LDS with the
// (k, k+1) pair interleaved so each B fragment is a single ds_load_b64
// landing directly in an even-aligned VGPR pair (no repack movs).
//   lds_idx(k, col) = (k>>1)*128 + col*2 + (k&1)
// ---------------------------------------------------------------------------
__global__ __launch_bounds__(128)
void dirsage_gemm_kernel(const float* __restrict__ x,
                         const float* __restrict__ sum_in,
                         const float* __restrict__ sum_out,
                         const float* __restrict__ cnt_in,
                         const float* __restrict__ cnt_out,
                         const float* __restrict__ Wself,
                         const float* __restrict__ bself,
                         const float* __restrict__ Ws2d,
                         const float* __restrict__ bs2d,
                         const float* __restrict__ Wd2s,
                         const float* __restrict__ bd2s,
                         float* __restrict__ out,
                         int N) {
    __shared__ float wl[3 * D * D];   // 48 KB of 320 KB LDS

    // Stage the three 64x64 weight matrices into LDS, k-pair interleaved.
    for (int i = threadIdx.x; i < D * D; i += blockDim.x) {
        int k   = i >> 6;
        int col = i & 63;
        int dst = ((k >> 1) << 7) + (col << 1) + (k & 1);
        wl[dst]             = Wself[i];
        wl[D * D + dst]     = Ws2d[i];
        wl[2 * D * D + dst] = Wd2s[i];
    }
    __syncthreads();

    int wave  = threadIdx.x >> 5;     // wave32
    int lane  = threadIdx.x & 31;
    int mtile = blockIdx.x * (blockDim.x >> 5) + wave;
    if (mtile * 16 >= N) return;      // whole wave exits together

    int row   = lane & 15;            // A: M index / B: N index within tile
    int half  = lane >> 4;            // 0: K={0,1}, 1: K={2,3}
    int khalf = half * 2;
    int gr    = mtile * 16 + row;     // global row this lane loads A for

    // Fold mean-normalization and alpha weighting into the A fragments.
    float sc_in  = (1.0f - ALPHA) / fmaxf(cnt_in[gr], 1.0f);
    float sc_out = ALPHA          / fmaxf(cnt_out[gr], 1.0f);

    // Combined bias initializes the C accumulators (bias depends on N only;
    // in C/D layout the N index within a tile is lane&15 for every VGPR).
    v8f acc[4];
    for (int t = 0; t < 4; ++t) {
        float b = bself[t * 16 + row]
                + (1.0f - ALPHA) * bs2d[t * 16 + row]
                + ALPHA * bd2s[t * 16 + row];
        for (int r = 0; r < 8; ++r) acc[t][r] = b;
    }

    const float* xrow = x       + (long long)gr * D;
    const float* irow = sum_in  + (long long)gr * D;
    const float* orow = sum_out + (long long)gr * D;

    // Per-lane LDS base: half selects K-pair (0 or 1 -> +128 floats per pair
    // row), row*2 is the interleaved column offset.
    const float* wbase = &wl[half * 128 + row * 2];

    for (int k0 = 0; k0 < D; k0 += 4) {
        int ka = k0 + khalf;          // 8-byte aligned (khalf even)
        // A fragments: 16x4 f32, one float2 (global_load_b64) per lane.
        v2f ax = *(const v2f*)(xrow + ka);
        v2f ai = *(const v2f*)(irow + ka);
        v2f ao = *(const v2f*)(orow + ka);
        ai *= sc_in;
        ao *= sc_out;

        const float* wk = wbase + (k0 >> 1) * 128;  // advances 2 pair-rows/step
        for (int t = 0; t < 4; ++t) {
            // B fragments: one ds_load_b64 each, immediate offsets.
            v2f bs = *(const v2f*)(wk + t * 32);
            v2f bi = *(const v2f*)(wk + D * D + t * 32);
            v2f bo = *(const v2f*)(wk + 2 * D * D + t * 32);

            acc[t] = __builtin_amdgcn_wmma_f32_16x16x4_f32(
                false, ax, false, bs, (short)0, acc[t], false, false);
            acc[t] = __builtin_amdgcn_wmma_f32_16x16x4_f32(
                false, ai, false, bi, (short)0, acc[t], false, false);
            acc[t] = __builtin_amdgcn_wmma_f32_16x16x4_f32(
                false, ao, false, bo, (short)0, acc[t], false, false);
        }
    }

    // C/D layout: VGPR r -> (M = r + 8*(lane>>4), N = lane&15) within tile.
    int mofs = half * 8;
    for (int t = 0; t < 4; ++t) {
        int col = t * 16 + row;
        long long base = (long long)(mtile * 16 + mofs) * D + col;
        for (int r = 0; r < 8; ++r) {
            out[base + (long long)r * D] = acc[t][r];
        }
    }
}

// ---------------------------------------------------------------------------
// Host launcher
// ---------------------------------------------------------------------------
extern "C" void kernel_launch(void* const* d_in, const int* in_sizes, int n_in,
                              void* d_out, int out_size, void* d_ws, size_t ws_size,
                              hipStream_t stream) {
    const float* x     = (const float*)d_in[0];
    const int*   ei    = (const int*)  d_in[1];
    const float* Wself = (const float*)d_in[2];
    const float* bself = (const float*)d_in[3];
    const float* Ws2d  = (const float*)d_in[4];
    const float* bs2d  = (const float*)d_in[5];
    const float* Wd2s  = (const float*)d_in[6];
    const float* bd2s  = (const float*)d_in[7];
    float* out = (float*)d_out;

    int N = in_sizes[0] / D;      // 100000
    int E = in_sizes[1] / 2;      // 1600000

    // Workspace layout: sum_in[N*D] | sum_out[N*D] | cnt_in[N] | cnt_out[N]
    float* ws      = (float*)d_ws;
    float* sum_in  = ws;
    float* sum_out = ws + (size_t)N * D;
    float* cnt_in  = ws + 2 * (size_t)N * D;
    float* cnt_out = cnt_in + N;

    int zn = 2 * N * D + 2 * N;
    zero_ws_kernel<<<(zn + 255) / 256, 256, 0, stream>>>(ws, zn);

    long long sthreads = (long long)E * (D / 4);
    int sblocks = (int)((sthreads + 255) / 256);
    scatter_edges_kernel<<<sblocks, 256, 0, stream>>>(
        x, ei, sum_in, sum_out, cnt_in, cnt_out, E);

    int mtiles = (N + 15) / 16;   // 6250
    int wavesPerBlock = 4;
    int gblocks = (mtiles + wavesPerBlock - 1) / wavesPerBlock;
    dirsage_gemm_kernel<<<gblocks, 32 * wavesPerBlock, 0, stream>>>(
        x, sum_in, sum_out, cnt_in, cnt_out,
        Wself, bself, Ws2d, bs2d, Wd2s, bd2s, out, N);
}